// TextEmbedding_35905926595499
// MI455X (gfx1250) — compile-verified
//
#include <hip/hip_runtime.h>
#include <hip/hip_bf16.h>

// Problem constants (from the reference).
constexpr int B_    = 32;
constexpr int NT_   = 1024;
constexpr int S_    = 4096;
constexpr int D_    = 512;

typedef float v4f __attribute__((ext_vector_type(4)));

// ---------------------------------------------------------------------------
// Kernel: length-regulator upsample + embedding gather.
// Block (128, 4): threadIdx.x indexes D in float4 chunks (128 * 4 = 512),
// threadIdx.y selects one of 4 output positions. Grid (S/4, B).
//
// A wave32 spans lanes with identical threadIdx.y, so `t` is wave-uniform:
// we readfirstlane it so the whole repeat-schedule division chain, the
// text[b,j] fetch, and the W_embed row base land in SGPRs. Each lane then
// does one b128 gather (temporal; table is L2-resident) and one b128
// NON-TEMPORAL store into the 256MB streaming output.
// ---------------------------------------------------------------------------
__global__ void upsample_embed_kernel(const int*   __restrict__ text,
                                      const int*   __restrict__ seq_len,
                                      const float* __restrict__ W_embed,
                                      const int*   __restrict__ lens,
                                      float*       __restrict__ out) {
    const int b = blockIdx.y;
    // Wave-uniform output position (lanes in a wave share threadIdx.y).
    const int t = __builtin_amdgcn_readfirstlane(blockIdx.x * 4 + threadIdx.y);

    const int A = seq_len[b];                     // uniform -> SGPR
    const int L = lens[b];                        // uniform -> SGPR

    v4f v = (v4f)(0.0f, 0.0f, 0.0f, 0.0f);

    if (t < A && L > 0 && A > 0) {
        // Repeat schedule: base = A // L, r = A % L,
        // split = (L - r) * base; positions < split map in blocks of `base`,
        // the rest in blocks of (base + 1).
        const int base  = A / L;
        const int r     = A - base * L;
        const int split = (L - r) * base;

        int j;
        if (t < split) {
            j = t / max(base, 1);
        } else {
            j = (L - r) + (t - split) / (base + 1);
        }
        j = min(max(j, 0), NT_ - 1);              // mathematically j <= L-1

        const int id = text[b * NT_ + j] + 1;     // pads (-1) -> row 0; scalar
        const v4f* __restrict__ row =
            (const v4f*)(W_embed + (size_t)id * D_);
        v = row[threadIdx.x];                     // global_load_b128, SGPR base
    }

    v4f* __restrict__ o =
        (v4f*)(out + (size_t)b * S_ * D_ + (size_t)t * D_);
    // Streaming 256MB output: non-temporal store keeps the ~5MB embedding
    // table resident in the 192MB L2 instead of being flushed by the stream.
    __builtin_nontemporal_store(v, &o[threadIdx.x]);
}

// ---------------------------------------------------------------------------
// Kernel: per-sample valid-token count L[b].
// L[b] = #{ j : j < min(NT, seq_len[b]) and text[b,j] != -1 }
// One block per sample; fully unrolled 4 loads/thread + LDS tree reduction.
// Workspace is rewritten every call (graph-replay safe, no persistent state).
// ---------------------------------------------------------------------------
__global__ void text_len_kernel(const int* __restrict__ text,
                                const int* __restrict__ seq_len,
                                int* __restrict__ lens) {
    __shared__ int sred[256];
    const int b   = blockIdx.x;
    const int lim = min(NT_, seq_len[b]);

    int c = 0;
#pragma unroll
    for (int k = 0; k < NT_ / 256; ++k) {
        const int j = (int)threadIdx.x + k * 256;     // always < NT_
        const int w = text[b * NT_ + j];
        c += (j < lim && w != -1) ? 1 : 0;
    }
    sred[threadIdx.x] = c;
    __syncthreads();
    for (int s = 128; s > 0; s >>= 1) {
        if ((int)threadIdx.x < s) sred[threadIdx.x] += sred[threadIdx.x + s];
        __syncthreads();
    }
    if (threadIdx.x == 0) lens[b] = sred[0];
}

extern "C" void kernel_launch(void* const* d_in, const int* in_sizes, int n_in,
                              void* d_out, int out_size, void* d_ws, size_t ws_size,
                              hipStream_t stream) {
    const int*   text    = (const int*)  d_in[0];   // [B, NT] int32
    const int*   seq_len = (const int*)  d_in[1];   // [B]     int32
    const float* W_embed = (const float*)d_in[2];   // [VOCAB+1, D] float32
    float*       out     = (float*)d_out;           // [B, S, D] float32
    int*         lens    = (int*)d_ws;              // B ints of scratch

    (void)in_sizes; (void)n_in; (void)out_size; (void)ws_size;

    text_len_kernel<<<dim3(B_), dim3(256), 0, stream>>>(text, seq_len, lens);

    dim3 grid(S_ / 4, B_);
    dim3 block(128, 4);
    upsample_embed_kernel<<<grid, block, 0, stream>>>(text, seq_len, W_embed,
                                                      lens, out);
}